// WCPatchLoss_22308060136254
// MI455X (gfx1250) — compile-verified
//
#include <hip/hip_runtime.h>
#include <hip/hip_bf16.h>

// MI455X / gfx1250, wave32. HBM-bound streaming reduction:
//  - kernel 1: one workgroup (8 wave32s) per 64x64 patch; Haar-of-diff +
//    charbonnier, register-resident channel sums (deterministic, no atomics),
//    hierarchical shuffle/LDS reduction -> 3 partials per patch in d_ws.
//  - kernel 2: one wave; sums the 3x1024 partials with V_WMMA_F32_16X16X4_F32
//    (ones-matrix trick, full fp32 precision), writes the final scalar.

typedef float v2f __attribute__((ext_vector_type(2)));
typedef float v8f __attribute__((ext_vector_type(8)));

#define EPS_F 1e-12f

__device__ __forceinline__ float charb(float v) {
    return sqrtf(fmaf(v, v, EPS_F));
}

__device__ __forceinline__ float waveRedSum(float v) {
#pragma unroll
    for (int m = 16; m >= 1; m >>= 1) v += __shfl_xor(v, m, 32);
    return v;
}

__device__ __forceinline__ float waveRedMax(float v) {
#pragma unroll
    for (int m = 16; m >= 1; m >>= 1) v = fmaxf(v, __shfl_xor(v, m, 32));
    return v;
}

// ---------------------------------------------------------------------------
// Kernel 1: per-patch streaming pass.
// Patch n (0..1023): channels c'=0..2 map to N=3n+c', plane t=N>>4 (192 planes
// of 256x256), quadrant q=N&15 -> (i,j)=(q>>2,q&3). 2x2 block (yy,xx):
// rows i*64+2yy(+1), cols j*64+2xx(+1).
// Thread map: wave w (0..7), lane l (0..31): yy = 4w+r (r=0..3), xx = l.
// ---------------------------------------------------------------------------
__global__ __launch_bounds__(256) void patch_main_kernel(
    const float* __restrict__ pred, const float* __restrict__ targ,
    float* __restrict__ part_band, float* __restrict__ part_rgb,
    float* __restrict__ part_max) {
    const int n    = blockIdx.x;      // patch id
    const int tid  = threadIdx.x;
    const int w    = tid >> 5;        // wave in workgroup
    const int lane = tid & 31;        // == xx

    float sum_band = 0.0f;            // sum over (c',yy,xx) of (lLH+lHL+lHH)
    float sum_rgb  = 0.0f;            // sum over pixels of sqrt(d^2+eps)
    float lmax     = 0.0f;            // max over this thread's (yy,xx) of channel sums

#pragma unroll
    for (int r = 0; r < 4; ++r) {
        const int yy = (w << 2) + r;
        float csum = 0.0f;            // sum over c' at this (yy,xx)
#pragma unroll
        for (int c = 0; c < 3; ++c) {
            const int N    = 3 * n + c;
            const int t    = N >> 4;
            const int q    = N & 15;
            const int rowY = ((q >> 2) << 6) + (yy << 1);
            const int colX = ((q & 3) << 6) + (lane << 1);
            const size_t base = (size_t)t * 65536u + (size_t)rowY * 256u + (size_t)colX;

            // 32 lanes x float2 = one contiguous 256B segment per row -> b64 loads
            const float2 p0 = *(const float2*)(pred + base);
            const float2 p1 = *(const float2*)(pred + base + 256);
            const float2 t0 = *(const float2*)(targ + base);
            const float2 t1 = *(const float2*)(targ + base + 256);

            const float a  = p0.x - t0.x;   // top-left
            const float b  = p0.y - t0.y;   // top-right
            const float cc = p1.x - t1.x;   // bottom-left
            const float d  = p1.y - t1.y;   // bottom-right

            // Haar is linear: band diffs = Haar of pixel diffs
            const float lLH = 0.5f * (a + b - cc - d);
            const float lHL = 0.5f * (a - b + cc - d);
            const float lHH = 0.5f * (a - b - cc + d);

            const float bs = charb(lLH) + charb(lHL) + charb(lHH);
            sum_band += bs;
            csum     += bs;
            sum_rgb  += charb(a) + charb(b) + charb(cc) + charb(d);
        }
        lmax = fmaxf(lmax, csum);
    }

    // workgroup reduction: wave shuffles + tiny LDS
    __shared__ float redB[8], redR[8], redM[8];
    const float vb = waveRedSum(sum_band);
    const float vr = waveRedSum(sum_rgb);
    const float vm = waveRedMax(lmax);
    if (lane == 0) { redB[w] = vb; redR[w] = vr; redM[w] = vm; }
    __syncthreads();
    if (tid == 0) {
        float tb = 0.0f, tr = 0.0f, tm = 0.0f;
#pragma unroll
        for (int i = 0; i < 8; ++i) {
            tb += redB[i];
            tr += redR[i];
            tm = fmaxf(tm, redM[i]);
        }
        part_band[n] = tb;
        part_rgb[n]  = tr;
        part_max[n]  = tm * (1.0f / 3.0f);   // mean over the 3 channels
    }
}

// ---------------------------------------------------------------------------
// Kernel 2: single wave32. Sum the three 1024-element partial arrays with
// V_WMMA_F32_16X16X4_F32: D = A(16x4) * ones(4x16) + C accumulates 64 values
// per instruction. Any bijective packing of data into A is sum-correct, and
// summing all 256 entries of D and dividing by 16 is layout-robust.
// EXEC is all-1s (32 uniform threads) as WMMA requires.
// ---------------------------------------------------------------------------
__global__ __launch_bounds__(32) void patch_final_kernel(
    const float* __restrict__ part_band, const float* __restrict__ part_rgb,
    const float* __restrict__ part_max, float* __restrict__ out) {
    const int lane = threadIdx.x;

    v8f accB = {0.f, 0.f, 0.f, 0.f, 0.f, 0.f, 0.f, 0.f};
    v8f accR = {0.f, 0.f, 0.f, 0.f, 0.f, 0.f, 0.f, 0.f};
    v8f accM = {0.f, 0.f, 0.f, 0.f, 0.f, 0.f, 0.f, 0.f};
    v2f ones; ones.x = 1.0f; ones.y = 1.0f;

    for (int c = 0; c < 16; ++c) {
        const int i0 = c * 64 + lane * 2;
        v2f aB; aB.x = part_band[i0]; aB.y = part_band[i0 + 1];
        v2f aR; aR.x = part_rgb[i0];  aR.y = part_rgb[i0 + 1];
        v2f aM; aM.x = part_max[i0];  aM.y = part_max[i0 + 1];
        accB = __builtin_amdgcn_wmma_f32_16x16x4_f32(
            false, aB, false, ones, (short)0, accB, false, false);
        accR = __builtin_amdgcn_wmma_f32_16x16x4_f32(
            false, aR, false, ones, (short)0, accR, false, false);
        accM = __builtin_amdgcn_wmma_f32_16x16x4_f32(
            false, aM, false, ones, (short)0, accM, false, false);
    }

    float sB = 0.0f, sR = 0.0f, sM = 0.0f;
#pragma unroll
    for (int i = 0; i < 8; ++i) { sB += accB[i]; sR += accR[i]; sM += accM[i]; }
    // each matrix total is replicated across the 16 columns of D
    sB = waveRedSum(sB) * (1.0f / 16.0f);
    sR = waveRedSum(sR) * (1.0f / 16.0f);
    sM = waveRedSum(sM) * (1.0f / 16.0f);

    if (lane == 0) {
        const float band_mean = sB / 3145728.0f;   // 1024*3*32*32
        const float max_mean  = sM / 1024.0f;      // mean over patches
        const float rgb_mean  = sR / 12582912.0f;  // 4*16*3*256*256
        out[0] = 0.5f * (band_mean + max_mean) + 0.5f * rgb_mean;
    }
}

extern "C" void kernel_launch(void* const* d_in, const int* in_sizes, int n_in,
                              void* d_out, int out_size, void* d_ws, size_t ws_size,
                              hipStream_t stream) {
    const float* pred = (const float*)d_in[0];
    const float* targ = (const float*)d_in[1];
    float* part_band = (float*)d_ws;          // 1024 floats
    float* part_rgb  = part_band + 1024;      // 1024 floats
    float* part_max  = part_band + 2048;      // 1024 floats (12 KB total)

    patch_main_kernel<<<1024, 256, 0, stream>>>(pred, targ, part_band, part_rgb, part_max);
    patch_final_kernel<<<1, 32, 0, stream>>>(part_band, part_rgb, part_max, (float*)d_out);
}